// NGCF_84164179132780
// MI455X (gfx1250) — compile-verified
//
#include <hip/hip_runtime.h>
#include <hip/hip_bf16.h>

typedef __attribute__((ext_vector_type(2))) float v2f;
typedef __attribute__((ext_vector_type(8))) float v8f;

#define EMB 64
#define TWAVES 5   // waves (16-row tiles) per transform block

// ---------------- init: ego = concat(user_emb, item_emb) ----------------
__global__ void init_ego_kernel(const float* __restrict__ ue, const float* __restrict__ ie,
                                float* __restrict__ ego, long long n_user_f4, long long total_f4) {
    long long i = (long long)blockIdx.x * blockDim.x + threadIdx.x;
    if (i >= total_f4) return;
    float4 v = (i < n_user_f4) ? ((const float4*)ue)[i] : ((const float4*)ie)[i - n_user_f4];
    ((float4*)ego)[i] = v;
}

// ---------------- zero side ----------------
__global__ void zero_kernel(float* __restrict__ p, long long n_f4) {
    long long i = (long long)blockIdx.x * blockDim.x + threadIdx.x;
    if (i >= n_f4) return;
    ((float4*)p)[i] = make_float4(0.f, 0.f, 0.f, 0.f);
}

// ---------------- scatter: side[dst] += ego[src] * w  (16 threads/edge, float4 each) ----------------
__global__ void scatter_kernel(const float* __restrict__ ego, const float* __restrict__ ew,
                               const int* __restrict__ src, const int* __restrict__ dst,
                               float* __restrict__ side, long long n_edge) {
    long long gid = (long long)blockIdx.x * blockDim.x + threadIdx.x;
    long long e = gid >> 4;
    if (e >= n_edge) return;
    int d4 = ((int)gid & 15) << 2;
    long long s = src[e], t = dst[e];
    float w = ew[e];
    float4 v = *(const float4*)(ego + s * EMB + d4);
    float* p = side + t * EMB + d4;
    atomicAdd(p + 0, v.x * w);
    atomicAdd(p + 1, v.y * w);
    atomicAdd(p + 2, v.z * w);
    atomicAdd(p + 3, v.w * w);
}

// ---------------- transform: ego = leakyrelu(side@Wg + bg + (ego*side)@Wb + bb) in place ----------------
// One wave per 16-row tile, V_WMMA_F32_16X16X4_F32, K=64 in 16 steps, 4 N-tiles, 2 GEMMs fused.
__global__ __launch_bounds__(TWAVES * 32)
void transform_kernel(float* __restrict__ ego, const float* __restrict__ side,
                      const float* __restrict__ Wg, const float* __restrict__ bg,
                      const float* __restrict__ Wb, const float* __restrict__ bb,
                      long long N) {
    // per-wave staged tiles: S (side rows) and P (ego*side rows), rows padded to 68 floats
    __shared__ float lds[TWAVES][2][16 * 68];
    const int tid  = threadIdx.x;
    const int wave = tid >> 5;
    const int lane = tid & 31;
    const int m  = lane & 15;   // row within tile (A) / col within N-tile (B,C)
    const int kh = lane >> 4;   // K-half selector
    const long long base = ((long long)blockIdx.x * TWAVES + wave) * 16;

    float* Sl = &lds[wave][0][0];
    float* Pl = &lds[wave][1][0];

    // stage 16x64 S and P tiles (coalesced float4 loads)
    #pragma unroll
    for (int i = 0; i < 8; ++i) {
        int f4  = i * 32 + lane;          // 0..255 float4 slots
        int row = f4 >> 4;                // 0..15
        int c4  = (f4 & 15) << 2;         // 0..60
        long long gr = base + row;
        if (gr >= N) gr = N - 1;          // clamp (N divides evenly here; safety only)
        float4 sv = *(const float4*)(side + gr * EMB + c4);
        float4 ev = *(const float4*)(ego  + gr * EMB + c4);
        *(float4*)(Sl + row * 68 + c4) = sv;
        float4 pv = make_float4(sv.x * ev.x, sv.y * ev.y, sv.z * ev.z, sv.w * ev.w);
        *(float4*)(Pl + row * 68 + c4) = pv;
    }
    __syncthreads();

    // accumulators initialized with broadcast bias (bias depends only on N index)
    v8f acc[4];
    #pragma unroll
    for (int t = 0; t < 4; ++t) {
        float bv = bg[t * 16 + m] + bb[t * 16 + m];
        #pragma unroll
        for (int r = 0; r < 8; ++r) acc[t][r] = bv;
    }

    #pragma unroll
    for (int kk = 0; kk < 16; ++kk) {
        const int krow = kk * 4 + 2 * kh;       // this lane's K rows: krow, krow+1
        v2f aS, aP;
        aS.x = Sl[m * 68 + krow];  aS.y = Sl[m * 68 + krow + 1];
        aP.x = Pl[m * 68 + krow];  aP.y = Pl[m * 68 + krow + 1];
        #pragma unroll
        for (int t = 0; t < 4; ++t) {
            const int n = t * 16 + m;
            v2f bG, bB;
            bG.x = Wg[(krow    ) * EMB + n];
            bG.y = Wg[(krow + 1) * EMB + n];
            bB.x = Wb[(krow    ) * EMB + n];
            bB.y = Wb[(krow + 1) * EMB + n];
            acc[t] = __builtin_amdgcn_wmma_f32_16x16x4_f32(false, aS, false, bG,
                                                           (short)0, acc[t], false, false);
            acc[t] = __builtin_amdgcn_wmma_f32_16x16x4_f32(false, aP, false, bB,
                                                           (short)0, acc[t], false, false);
        }
    }

    // leaky-relu(0.2) and in-place writeback (unnormalized -> next layer's ego)
    #pragma unroll
    for (int t = 0; t < 4; ++t) {
        #pragma unroll
        for (int r = 0; r < 8; ++r) {
            float v = acc[t][r];
            v = v > 0.f ? v : 0.2f * v;
            long long row = base + r + 8 * kh;
            if (row < N) ego[row * EMB + t * 16 + m] = v;
        }
    }
}

// ---------------- gather: one wave per output row; optional L2 row-normalize ----------------
__global__ void gather_kernel(const float* __restrict__ ego,
                              const int* __restrict__ users, const int* __restrict__ pos,
                              const int* __restrict__ neg,
                              float* __restrict__ out, int batch, long long n_user,
                              int col_off, int cols_total, int do_norm) {
    int gid  = blockIdx.x * blockDim.x + threadIdx.x;
    int row  = gid >> 5;
    int lane = gid & 31;
    if (row >= 3 * batch) return;
    int which = row / batch;
    int b     = row - which * batch;
    long long node = (which == 0) ? (long long)users[b]
                   : (which == 1) ? n_user + (long long)pos[b]
                                  : n_user + (long long)neg[b];
    float2 v = *(const float2*)(ego + node * EMB + lane * 2);
    float scale = 1.f;
    if (do_norm) {
        float ss = v.x * v.x + v.y * v.y;
        #pragma unroll
        for (int off = 16; off > 0; off >>= 1) ss += __shfl_xor(ss, off, 32);
        scale = 1.f / fmaxf(sqrtf(ss), 1e-12f);
    }
    float2 o = make_float2(v.x * scale, v.y * scale);
    *(float2*)(out + ((long long)which * batch + b) * cols_total + col_off + lane * 2) = o;
}

extern "C" void kernel_launch(void* const* d_in, const int* in_sizes, int n_in,
                              void* d_out, int out_size, void* d_ws, size_t ws_size,
                              hipStream_t stream) {
    (void)n_in; (void)out_size; (void)ws_size;
    const float* user_emb = (const float*)d_in[0];
    const float* item_emb = (const float*)d_in[1];
    const float* W_gc     = (const float*)d_in[2];
    const float* b_gc     = (const float*)d_in[3];
    const float* W_bi     = (const float*)d_in[4];
    const float* b_bi     = (const float*)d_in[5];
    const float* edge_w   = (const float*)d_in[6];
    const int*   edge_src = (const int*)d_in[7];
    const int*   edge_dst = (const int*)d_in[8];
    const int*   users    = (const int*)d_in[9];
    const int*   pos      = (const int*)d_in[10];
    const int*   neg      = (const int*)d_in[11];

    const long long n_user = in_sizes[0] / EMB;
    const long long n_item = in_sizes[1] / EMB;
    const long long N      = n_user + n_item;
    const int layers       = in_sizes[2] / (EMB * EMB);
    const long long n_edge = in_sizes[6];
    const int batch        = in_sizes[9];
    const int cols_total   = (layers + 1) * EMB;

    float* ego  = (float*)d_ws;                 // N x 64
    float* side = ego + (size_t)N * EMB;        // N x 64

    const long long n_f4 = N * EMB / 4;
    init_ego_kernel<<<(int)((n_f4 + 255) / 256), 256, 0, stream>>>(
        user_emb, item_emb, ego, n_user * EMB / 4, n_f4);

    // layer-0 slice of all_e is the raw embeddings (unnormalized)
    {
        int threads = 3 * batch * 32;
        gather_kernel<<<(threads + 255) / 256, 256, 0, stream>>>(
            ego, users, pos, neg, (float*)d_out, batch, n_user, 0, cols_total, 0);
    }

    const int tile_blocks = (int)((N + (long long)TWAVES * 16 - 1) / (TWAVES * 16)); // 1875

    for (int k = 0; k < layers; ++k) {
        zero_kernel<<<(int)((n_f4 + 255) / 256), 256, 0, stream>>>(side, n_f4);

        long long sthreads = n_edge * 16;
        scatter_kernel<<<(int)((sthreads + 255) / 256), 256, 0, stream>>>(
            ego, edge_w, edge_src, edge_dst, side, n_edge);

        transform_kernel<<<tile_blocks, TWAVES * 32, 0, stream>>>(
            ego, side,
            W_gc + (size_t)k * EMB * EMB, b_gc + (size_t)k * EMB,
            W_bi + (size_t)k * EMB * EMB, b_bi + (size_t)k * EMB, N);

        int threads = 3 * batch * 32;
        gather_kernel<<<(threads + 255) / 256, 256, 0, stream>>>(
            ego, users, pos, neg, (float*)d_out, batch, n_user, (k + 1) * EMB, cols_total, 1);
    }
}